// ContrastiveLoss_21698174779780
// MI455X (gfx1250) — compile-verified
//
#include <hip/hip_runtime.h>
#include <hip/hip_bf16.h>
#include <math.h>

typedef __attribute__((ext_vector_type(2))) float v2f;
typedef __attribute__((ext_vector_type(8))) float v8f;
typedef __attribute__((ext_vector_type(4))) unsigned int u32x4;
typedef __attribute__((ext_vector_type(8))) int i32x8;
typedef __attribute__((ext_vector_type(4))) int i32x4;

// -------------------------------------------------------------------------
// Tensor Data Mover: DMA a 1-D run of D f32 elements from global -> LDS.
// D# per CDNA5 ISA: group0 = {count=1 | lds_addr | global_addr | type=2},
// group1 = {data_size=4B, tensor_dim0=D, tensor_dim1=1, tile_dim0=D,
//           tile_dim1=1, tensor_dim0_stride=D}.  Wave-uniform operands.
// -------------------------------------------------------------------------
__device__ __forceinline__ void tdm_load_1d_to_lds(const float* src,
                                                   unsigned lds_byte_off, int D)
{
    unsigned long long ga = (unsigned long long)(uintptr_t)src;
    u32x4 g0;
    g0[0] = 1u;                                   // count=1, user descriptor
    g0[1] = lds_byte_off;                         // lds_addr (bytes)
    g0[2] = (unsigned)(ga & 0xFFFFFFFFu);         // global_addr[31:0]
    g0[3] = (unsigned)((ga >> 32) & 0x1FFFFFFu)   // global_addr[56:32]
          | (2u << 30);                           // type = 2 ("image")
    i32x8 g1;
    g1[0] = 0x20000;                              // data_size = 2 (4 bytes)
    g1[1] = (D & 0xFFFF) << 16;                   // tensor_dim0 [15:0]
    g1[2] = ((D >> 16) & 0xFFFF) | (1 << 16);     // tensor_dim0[31:16], tensor_dim1=1
    g1[3] = (D & 0xFFFF) << 16;                   // tile_dim0 = D
    g1[4] = 1;                                    // tile_dim1 = 1, tile_dim2 = 0
    g1[5] = D;                                    // tensor_dim0_stride[31:0]
    g1[6] = 0;
    g1[7] = 0;
    i32x4 g2 = {0, 0, 0, 0};
    i32x4 g3 = {0, 0, 0, 0};
#if __clang_major__ >= 23
    i32x8 gx = {0, 0, 0, 0, 0, 0, 0, 0};
    __builtin_amdgcn_tensor_load_to_lds(g0, g1, g2, g3, gx, 0);
#else
    __builtin_amdgcn_tensor_load_to_lds(g0, g1, g2, g3, 0);
#endif
}

// -------------------------------------------------------------------------
// K0: norms of the two reference vectors: norms[0]=||v2||, norms[1]=||v3||
// -------------------------------------------------------------------------
__global__ __launch_bounds__(256) void vecnorm_kernel(
    const float* __restrict__ v2, const float* __restrict__ v3,
    float* __restrict__ norms, int D)
{
    __shared__ float red[256];
    int t = threadIdx.x;
    float s2 = 0.f, s3 = 0.f;
    for (int i = t; i < D; i += 256) {
        float a = v2[i]; s2 = fmaf(a, a, s2);
        float b = v3[i]; s3 = fmaf(b, b, s3);
    }
    red[t] = s2; __syncthreads();
    for (int off = 128; off > 0; off >>= 1) { if (t < off) red[t] += red[t + off]; __syncthreads(); }
    if (t == 0) norms[0] = sqrtf(red[0]);
    __syncthreads();
    red[t] = s3; __syncthreads();
    for (int off = 128; off > 0; off >>= 1) { if (t < off) red[t] += red[t + off]; __syncthreads(); }
    if (t == 0) norms[1] = sqrtf(red[0]);
}

// -------------------------------------------------------------------------
// K1: WMMA pass. Each wave owns a 16-row tile of A [N,D]:
//   dot(row, v2) / dot(row, v3) via V_WMMA_F32_16X16X4_F32, row sum-of-
//   squares via VALU fma (co-executes with the matrix pipe).
//
// A f32 16x4 layout: lanes 0-15 -> M=0..15 hold K=k0,k0+1; lanes 16-31 same
// rows hold K=k0+2,k0+3.  B 4x16 mirrors with N across lanes; col0=v2,
// col1=v3, cols 2..15 = 0.  B is fetched branch-free: every lane owns an LDS
// pointer (v2 chunk / v3 chunk / zero pad) advanced by a per-lane stride.
// C f32 16x16: column n -> lane n (M=0..7 in c[0..7]) and lane n+16 (M=8..15).
// -------------------------------------------------------------------------
__global__ __launch_bounds__(256) void dot_wmma_kernel(
    const float* __restrict__ A,
    const float* __restrict__ v2, const float* __restrict__ v3,
    float* __restrict__ dot2, float* __restrict__ dot3,
    float* __restrict__ sumsq, int N, int D)
{
    // [0..D) = v2, [D..2D) = v3, [2D..2D+8) = zero pad
    __shared__ float sB[2 * 512 + 8];

    if (threadIdx.x < 8) sB[2 * D + threadIdx.x] = 0.f;
    if (threadIdx.x < 32) {
        // wave 0 issues both TDM transfers, then drains TENSORcnt
        unsigned base = (unsigned)(uintptr_t)&sB[0];   // low 32 bits = LDS offset
        tdm_load_1d_to_lds(v2, base, D);
        tdm_load_1d_to_lds(v3, base + (unsigned)D * 4u, D);
        __builtin_amdgcn_s_wait_tensorcnt(0);
    }
    __syncthreads();

    const int lane = threadIdx.x & 31;
    const int wave = threadIdx.x >> 5;
    const int tile = blockIdx.x * 8 + wave;     // 16 rows per wave
    const int m0   = tile * 16;
    if (m0 >= N) return;
    const int half = lane >> 4;                 // 0: K pair lo, 1: K pair hi
    const int idx  = lane & 15;                 // row index (A) / col index (B)

    const float* rowp = A + (size_t)(m0 + idx) * (size_t)D + 2 * half;

    // branch-free per-lane B source
    int boff;
    if (idx == 0)      boff = 2 * half;
    else if (idx == 1) boff = D + 2 * half;
    else               boff = 2 * D;            // zero pad
    const float* bp  = sB + boff;
    const int bstep  = (idx < 2) ? 4 : 0;

    v8f   c  = {};
    float ss = 0.f;

    #pragma unroll 4
    for (int k0 = 0; k0 < D; k0 += 4) {
        if ((k0 & 127) == 0) {
            // speculative streaming prefetch of this row further ahead
            __builtin_prefetch(rowp + k0 + 512, 0, 1);
        }
        const v2f a = *reinterpret_cast<const v2f*>(rowp + k0);
        const v2f b = *reinterpret_cast<const v2f*>(bp);
        bp += bstep;
        ss = fmaf(a.x, a.x, ss);
        ss = fmaf(a.y, a.y, ss);
        c = __builtin_amdgcn_wmma_f32_16x16x4_f32(
                /*neg_a=*/false, a, /*neg_b=*/false, b,
                /*c_mod=*/(short)0, c, /*reuse_a=*/false, /*reuse_b=*/false);
    }

    // Row sum-of-squares: lane i holds the K-low half, lane i+16 the K-high.
    float row_ss = ss + __shfl_xor(ss, 16, 32);
    if (lane < 16) sumsq[m0 + idx] = row_ss;

    // Column 0 of C = dot(v2), column 1 = dot(v3): two b128 stores per lane.
    if (idx < 2) {
        float* dst = ((idx == 0) ? dot2 : dot3) + m0 + half * 8;
        float4 lo = make_float4(c[0], c[1], c[2], c[3]);
        float4 hi = make_float4(c[4], c[5], c[6], c[7]);
        *reinterpret_cast<float4*>(dst)     = lo;
        *reinterpret_cast<float4*>(dst + 4) = hi;
    }
}

// -------------------------------------------------------------------------
// K2: stable descending rank by counting (reproduces argsort(-r) stable
//     semantics), output ascending position q = N-1-pos_desc.
// -------------------------------------------------------------------------
__global__ __launch_bounds__(256) void rank_kernel(
    const float* __restrict__ r, int* __restrict__ q, int N)
{
    __shared__ float tile[256];
    const int i  = blockIdx.x * 256 + threadIdx.x;
    const float ri = r[i];
    int pos = 0;
    for (int base = 0; base < N; base += 256) {
        tile[threadIdx.x] = r[base + threadIdx.x];
        __syncthreads();
        #pragma unroll 8
        for (int j = 0; j < 256; ++j) {
            const float rj = tile[j];
            const int jj = base + j;
            pos += (int)((rj > ri) || ((rj == ri) && (jj < i)));
        }
        __syncthreads();
    }
    q[i] = (N - 1) - pos;   // ascending-rank position
}

// -------------------------------------------------------------------------
// K3: scatter exp(cosine) into ascending-rank order (forward cumsum of this
//     array == reference suffix sums).
// -------------------------------------------------------------------------
__global__ __launch_bounds__(256) void scatter_kernel(
    const float* __restrict__ dot2, const float* __restrict__ dot3,
    const float* __restrict__ sumsq, const int* __restrict__ q,
    const float* __restrict__ norms,
    float* __restrict__ E2, float* __restrict__ E3, int N)
{
    const int i = blockIdx.x * 256 + threadIdx.x;
    if (i >= N) return;
    const float nx = sqrtf(sumsq[i]);
    const float den2 = fmaxf(nx * norms[0], 1e-8f);
    const float den3 = fmaxf(nx * norms[1], 1e-8f);
    const int p = q[i];
    E2[p] = expf(dot2[i] / den2);
    E3[p] = expf(dot3[i] / den3);
}

// -------------------------------------------------------------------------
// K4: single-block chunked scan + loss reduction.
//     loss += sum_{p=0..N-2} ( log(total) - log(prefix_inclusive[p]) )
// -------------------------------------------------------------------------
__global__ __launch_bounds__(1024) void scan_loss_kernel(
    const float* __restrict__ E2, const float* __restrict__ E3,
    float* __restrict__ out, int N)
{
    constexpr int T = 1024;
    __shared__ float s[T];
    const int t = threadIdx.x;
    const int chunk = N / T;
    float loss = 0.f;

    for (int ch = 0; ch < 2; ++ch) {
        const float* E = (ch == 0) ? E2 : E3;
        const float* base = E + (size_t)t * chunk;

        float part = 0.f;
        for (int j = 0; j < chunk; ++j) part += base[j];
        s[t] = part;
        __syncthreads();

        // inclusive Hillis-Steele scan of the 1024 partials
        for (int off = 1; off < T; off <<= 1) {
            float v = (t >= off) ? s[t - off] : 0.f;
            __syncthreads();
            s[t] += v;
            __syncthreads();
        }
        const float excl  = s[t] - part;
        const float total = s[T - 1];
        const float logT  = logf(total);

        float run = excl;
        float acc = 0.f;
        for (int j = 0; j < chunk; ++j) {
            run += base[j];
            if (t * chunk + j < N - 1) acc += logT - logf(run);
        }
        __syncthreads();
        s[t] = acc;
        __syncthreads();
        for (int off = T >> 1; off > 0; off >>= 1) {
            if (t < off) s[t] += s[t + off];
            __syncthreads();
        }
        if (t == 0) loss += s[0];
        __syncthreads();
    }
    if (t == 0) out[0] = loss;
}

// -------------------------------------------------------------------------
extern "C" void kernel_launch(void* const* d_in, const int* in_sizes, int n_in,
                              void* d_out, int out_size, void* d_ws, size_t ws_size,
                              hipStream_t stream)
{
    const float* A  = (const float*)d_in[0];   // [N, D]
    const float* v2 = (const float*)d_in[1];   // [1, D]
    const float* v3 = (const float*)d_in[2];   // [1, D]
    const float* rk = (const float*)d_in[3];   // [N]
    const int D = in_sizes[1];
    const int N = in_sizes[3];

    float* w     = (float*)d_ws;
    float* dot2  = w;
    float* dot3  = w + (size_t)N;
    float* sumsq = w + 2 * (size_t)N;
    float* E2    = w + 3 * (size_t)N;
    float* E3    = w + 4 * (size_t)N;
    int*   q     = (int*)(w + 5 * (size_t)N);
    float* norms = w + 6 * (size_t)N;          // 2 floats

    vecnorm_kernel<<<1, 256, 0, stream>>>(v2, v3, norms, D);

    const int tiles  = N / 16;                 // 16 rows per wave
    const int blocks = tiles / 8;              // 8 waves per block
    dot_wmma_kernel<<<blocks, 256, 0, stream>>>(A, v2, v3, dot2, dot3, sumsq, N, D);

    rank_kernel<<<N / 256, 256, 0, stream>>>(rk, q, N);

    scatter_kernel<<<N / 256, 256, 0, stream>>>(dot2, dot3, sumsq, q, norms, E2, E3, N);

    scan_loss_kernel<<<1, 1024, 0, stream>>>(E2, E3, (float*)d_out, N);
}